// DifferentiableRaster_10419590660324
// MI455X (gfx1250) — compile-verified
//
#include <hip/hip_runtime.h>
#include <hip/hip_bf16.h>
#include <stdint.h>

#define HH 1024
#define WW 1024
#define BB 32
#define NN 262144

#define RBLK  32    // blocks per batch, min/max kernel
#define RBLK2 128   // blocks per batch, raster kernel

typedef float v2f __attribute__((ext_vector_type(2)));
typedef float v8f __attribute__((ext_vector_type(8)));

// Order-preserving map float -> uint32 (monotone for all finite floats).
__device__ __forceinline__ uint32_t mapf(float f) {
  uint32_t u = __float_as_uint(f);
  return u ^ ((u >> 31) ? 0xFFFFFFFFu : 0x80000000u);
}
__device__ __forceinline__ float unmapf(uint32_t u) {
  uint32_t b = (u & 0x80000000u) ? (u ^ 0x80000000u) : ~u;
  return __uint_as_float(b);
}

// Guaranteed single-instruction global_atomic_add_f32 (no CAS expansion).
// d_out is ordinary device memory, so fine-grained semantics are not needed.
__device__ __forceinline__ void atomAddF(float* p, float v) {
  unsafeAtomicAdd(p, v);
}

// ---------------------------------------------------------------------------
// Workspace layout (uint32 view):
//   [b*8 + 0..5] : mapped minx, maxx, miny, maxy, minz, maxz for batch b
//   [256 + b*2 + t] : mapped z order-statistic for rank (2621 + t)
// ---------------------------------------------------------------------------

// Kernel 1: zero 128MB output (L2-warming) + init workspace min/max slots.
__global__ void dr_zero_init(float4* __restrict__ out4, uint32_t* __restrict__ ws,
                             long n4) {
  long i = (long)blockIdx.x * blockDim.x + threadIdx.x;
  long stride = (long)gridDim.x * blockDim.x;
  for (long k = i; k < n4; k += stride) out4[k] = make_float4(0.f, 0.f, 0.f, 0.f);
  if (i < BB * 8) {
    int j = (int)i & 7;
    ws[i] = (j == 0 || j == 2 || j == 4) ? 0xFFFFFFFFu : 0u;  // mins=+max, maxs=0
  }
}

// Kernel 2: per-batch min/max of x, y, z (wave32 shuffle reduce + uint atomics).
__global__ void dr_minmax(const float* __restrict__ pc, uint32_t* __restrict__ ws) {
  const int b  = blockIdx.x / RBLK;
  const int cb = blockIdx.x % RBLK;
  const float* base = pc + (size_t)b * NN * 3;
  float mn[3] = {3.402823466e38f, 3.402823466e38f, 3.402823466e38f};
  float mx[3] = {-3.402823466e38f, -3.402823466e38f, -3.402823466e38f};
  for (int i = cb * blockDim.x + threadIdx.x; i < NN; i += RBLK * blockDim.x) {
    size_t o = 3 * (size_t)i;
    float v0 = base[o], v1 = base[o + 1], v2 = base[o + 2];
    mn[0] = fminf(mn[0], v0); mx[0] = fmaxf(mx[0], v0);
    mn[1] = fminf(mn[1], v1); mx[1] = fmaxf(mx[1], v1);
    mn[2] = fminf(mn[2], v2); mx[2] = fmaxf(mx[2], v2);
  }
  // wave32 reduction
  #pragma unroll
  for (int off = 16; off; off >>= 1) {
    #pragma unroll
    for (int c = 0; c < 3; ++c) {
      mn[c] = fminf(mn[c], __shfl_xor(mn[c], off, 32));
      mx[c] = fmaxf(mx[c], __shfl_xor(mx[c], off, 32));
    }
  }
  __shared__ float s_mn[3][8], s_mx[3][8];
  const int wid = threadIdx.x >> 5, lane = threadIdx.x & 31;
  if (lane == 0) {
    #pragma unroll
    for (int c = 0; c < 3; ++c) { s_mn[c][wid] = mn[c]; s_mx[c][wid] = mx[c]; }
  }
  __syncthreads();
  if (threadIdx.x == 0) {
    const int nw = blockDim.x >> 5;
    #pragma unroll
    for (int c = 0; c < 3; ++c) {
      float m = s_mn[c][0], M = s_mx[c][0];
      for (int w = 1; w < nw; ++w) { m = fminf(m, s_mn[c][w]); M = fmaxf(M, s_mx[c][w]); }
      atomicMin(&ws[b * 8 + 2 * c + 0], mapf(m));
      atomicMax(&ws[b * 8 + 2 * c + 1], mapf(M));
    }
  }
}

// Kernel 3: exact radix select of the rank-(2621+t) z value per batch.
// 4 passes of 8-bit digits (MSB first) on monotone-mapped bits.
// The 256-bin cumulative sum each pass runs on the matrix unit:
//   rowcum(16x16) = H(16x16) x U(16x16 upper-triangular ones)
// via 4 chained V_WMMA_F32_16X16X4_F32 (wave 0, EXEC all ones), then a
// 16-element block-offset scan. All counts <= 2^18, exact in f32.
__global__ void dr_radix_select(const float* __restrict__ pc, uint32_t* __restrict__ ws) {
  const int tid = threadIdx.x;
  const int b = blockIdx.x >> 1;
  const int t = blockIdx.x & 1;
  __shared__ uint32_t hist[256];
  __shared__ float histf[256];
  __shared__ float rowcum[256];
  __shared__ float cumf[256];
  __shared__ float Pofs[16];
  __shared__ uint32_t s_prefix, s_krem;
  const float* base = pc + (size_t)b * NN * 3;
  if (tid == 0) { s_prefix = 0u; s_krem = 2621u + (uint32_t)t; }
  __syncthreads();

  for (int pass = 3; pass >= 0; --pass) {
    for (int i = tid; i < 256; i += blockDim.x) hist[i] = 0u;
    __syncthreads();
    const uint32_t prefix = s_prefix;
    const uint32_t maskHi = (pass == 3) ? 0u : (0xFFFFFFFFu << (8 * (pass + 1)));
    const int shift = 8 * pass;

    for (int i = tid; i < NN; i += blockDim.x) {
      uint32_t u = mapf(base[3 * (size_t)i + 2]);
      if ((u & maskHi) == (prefix & maskHi))
        atomicAdd(&hist[(u >> shift) & 255u], 1u);
    }
    __syncthreads();
    for (int i = tid; i < 256; i += blockDim.x) histf[i] = (float)hist[i];
    __syncthreads();

    if (tid < 32) {  // wave 0, EXEC all ones: WMMA prefix-sum (fully uniform block)
      const int half = tid >> 4;   // K half select per the 32-bit A/B layouts
      const int l    = tid & 15;   // A row (M) == B column (N)
      v8f acc = {0.f, 0.f, 0.f, 0.f, 0.f, 0.f, 0.f, 0.f};
      #pragma unroll
      for (int j = 0; j < 4; ++j) {
        v2f a, bm;
        const int ka = 4 * j + 2 * half;
        a.x = histf[16 * l + ka];
        a.y = histf[16 * l + ka + 1];
        bm.x = (ka     <= l) ? 1.0f : 0.0f;   // U[k][n] = (k <= n)
        bm.y = (ka + 1 <= l) ? 1.0f : 0.0f;
        acc = __builtin_amdgcn_wmma_f32_16x16x4_f32(
            false, a, false, bm, (short)0, acc, false, false);
      }
      #pragma unroll
      for (int r = 0; r < 8; ++r) {
        const int row = r + 8 * half;          // C/D layout: VGPR r -> M = r + 8*(lane/16)
        rowcum[16 * row + l] = acc[r];
      }
    }
    __syncthreads();
    if (tid == 0) {
      float p = 0.f;
      #pragma unroll
      for (int m = 0; m < 16; ++m) { Pofs[m] = p; p += rowcum[16 * m + 15]; }
    }
    __syncthreads();
    for (int i = tid; i < 256; i += blockDim.x) cumf[i] = rowcum[i] + Pofs[i >> 4];
    __syncthreads();

    if (tid == 0) {
      const uint32_t krem = s_krem;
      int d = 255;
      uint32_t excl = 0u;
      for (int i = 0; i < 256; ++i) {
        const uint32_t incl = (uint32_t)(cumf[i] + 0.5f);
        if (krem < incl) {
          d = i;
          excl = (i == 0) ? 0u : (uint32_t)(cumf[i - 1] + 0.5f);
          break;
        }
      }
      s_krem = krem - excl;
      s_prefix = prefix | ((uint32_t)d << shift);
    }
    __syncthreads();
  }
  if (tid == 0) ws[256 + 2 * b + t] = s_prefix;
}

// Kernel 4: rasterize — 4 fp32 no-return atomic adds per point into the
// L2-resident (128MB < 192MB L2) image.
__global__ void dr_raster(const float* __restrict__ pc, const uint32_t* __restrict__ ws,
                          float* __restrict__ out) {
  const int b  = blockIdx.x / RBLK2;
  const int cb = blockIdx.x % RBLK2;
  const uint32_t* w = ws + b * 8;
  const float xmin = unmapf(w[0]), xmax = unmapf(w[1]);
  const float ymin = unmapf(w[2]), ymax = unmapf(w[3]);
  const float zmin = unmapf(w[4]), zmax = unmapf(w[5]);
  const float zlo = unmapf(ws[256 + 2 * b]), zhi = unmapf(ws[256 + 2 * b + 1]);
  const float hq = 0.43000000000000016f;     // 0.01*(N-1) - 2621
  const float q01z = zlo + hq * (zhi - zlo);
  const float rz = 1.0f / (zmax - zmin);
  const float q01 = (q01z - zmin) * rz;
  const float sx = (float)(HH - 2) / (xmax - xmin);
  const float sy = (float)(WW - 2) / (ymax - ymin);
  float* ob = out + (size_t)b * HH * WW;
  const float* base = pc + (size_t)b * NN * 3;

  for (int i = cb * blockDim.x + threadIdx.x; i < NN; i += RBLK2 * blockDim.x) {
    size_t o = 3 * (size_t)i;
    const float x = base[o], y = base[o + 1], z = base[o + 2];
    float dist = (z - zmin) * rz;
    dist = fmaxf(dist, q01);
    const float s = 1.0f - dist;
    const float xi = (x - xmin) * sx + 1.0f;   // row coord in [1, H-1]
    const float yi = (y - ymin) * sy + 1.0f;   // col coord in [1, W-1]
    const float f0 = floorf(xi), c0 = ceilf(xi);
    const float f1 = floorf(yi), c1 = ceilf(yi);
    const float dx = xi - f0, dy = yi - f1;    // >= 0
    const float ex = c0 - xi, ey = c1 - yi;    // >= 0
    const int r0 = (int)f0, r1 = (int)c0, q0 = (int)f1, q1 = (int)c1;
    // reference's (intentionally unconventional) corner weights:
    atomAddF(&ob[r0 * WW + q0], s * dx * dy);  // tl
    atomAddF(&ob[r1 * WW + q0], s * ex * dy);  // bl
    atomAddF(&ob[r0 * WW + q1], s * dx * ey);  // tr
    atomAddF(&ob[r1 * WW + q1], s * ex * ey);  // br
  }
}

extern "C" void kernel_launch(void* const* d_in, const int* in_sizes, int n_in,
                              void* d_out, int out_size, void* d_ws, size_t ws_size,
                              hipStream_t stream) {
  (void)in_sizes; (void)n_in; (void)ws_size;
  const float* pc = (const float*)d_in[0];   // [B, N, 3] float32
  float* out = (float*)d_out;                // [B, H, W] float32
  uint32_t* ws = (uint32_t*)d_ws;

  const long n4 = (long)out_size / 4;        // float4 count (out_size = 33,554,432)
  dr_zero_init<<<4096, 256, 0, stream>>>((float4*)out, ws, n4);
  dr_minmax<<<BB * RBLK, 256, 0, stream>>>(pc, ws);
  dr_radix_select<<<BB * 2, 256, 0, stream>>>(pc, ws);
  dr_raster<<<BB * RBLK2, 256, 0, stream>>>(pc, ws, out);
}